// DenseCondenser_41497974014509
// MI455X (gfx1250) — compile-verified
//
#include <hip/hip_runtime.h>

typedef __attribute__((ext_vector_type(2))) float v2f;
typedef __attribute__((ext_vector_type(8))) float v8f;
typedef __attribute__((ext_vector_type(4))) int   v4i;

typedef __attribute__((address_space(1))) v4i* gptr_b128;   // global, 16B granule
typedef __attribute__((address_space(3))) v4i* lptr_b128;   // LDS, 16B granule

#define KDIM 4096
#define NOUT 8
#define TM 128                    // samples per workgroup (8 waves * 16)
#define KC 64                     // K-chunk staged in LDS
#define NCHUNK (KDIM / KC)        // 64
#define WPC (KC / 4)              // 16 WMMAs per chunk per wave
#define LDSTRIDE 68               // 64 + 4 pad -> conflict-free A reads

#if defined(__gfx1250__) && __has_builtin(__builtin_amdgcn_global_load_async_to_lds_b128)
#define HAVE_ASYNC_LDS 1
#else
#define HAVE_ASYNC_LDS 0
#endif

__device__ __forceinline__ void wait_asynccnt0() {
#if __has_builtin(__builtin_amdgcn_s_wait_asynccnt)
  __builtin_amdgcn_s_wait_asynccnt(0);
#else
  asm volatile("s_wait_asynccnt 0x0" ::: "memory");
#endif
}

// ---------------------------------------------------------------------------
// Kernel 1: fold the three 8x8x8 tensor-train cores into the effective
// 4096x8 matrix Weff, stored pre-swizzled in the exact B-operand VGPR layout
// of v_wmma_f32_16x16x4_f32 (N padded to 16 with zeros):
//   Bp[wi*64 + lane*2 + comp]  where for k = wi*4 + off:
//     lane = (off>=2 ? 16:0) + n,  comp = off&1   (VGPR0 holds K0/K2, VGPR1 K1/K3)
// ---------------------------------------------------------------------------
__global__ void pack_weff(const float* __restrict__ n0,
                          const float* __restrict__ n1,
                          const float* __restrict__ n2,
                          float* __restrict__ Bp) {
  int k = blockIdx.x * blockDim.x + threadIdx.x;           // 0..4095
  int l4 = k & 7, l3 = (k >> 3) & 7, l2 = (k >> 6) & 7, l1 = (k >> 9) & 7;
  float a0[8], t[8], w[8];
#pragma unroll
  for (int c = 0; c < 8; ++c) a0[c] = n0[(l4 * 8 + l3) * 8 + c];       // node0[l4][l3][c]
#pragma unroll
  for (int c2 = 0; c2 < 8; ++c2) {
    float s = 0.f;
#pragma unroll
    for (int c = 0; c < 8; ++c) s += n1[(c * 8 + l2) * 8 + c2] * a0[c]; // node1[c][l2][c2]
    t[c2] = s;
  }
#pragma unroll
  for (int c3 = 0; c3 < 8; ++c3) {
    float s = 0.f;
#pragma unroll
    for (int c2 = 0; c2 < 8; ++c2) s += n2[(c2 * 8 + l1) * 8 + c3] * t[c2]; // node2[c2][l1][c3]
    w[c3] = s;
  }
  int wi = k >> 2, off = k & 3;
  int laneBase = (off & 2) ? 16 : 0;
  int comp = off & 1;
#pragma unroll
  for (int n = 0; n < 16; ++n) {
    float v = (n < NOUT) ? w[n & 7] : 0.0f;
    Bp[wi * 64 + (laneBase + n) * 2 + comp] = v;
  }
}

// ---------------------------------------------------------------------------
// Kernel 2: out(65536x8) = x(65536x4096) @ Weff(4096x8) + bias, via
// v_wmma_f32_16x16x4_f32. 128 samples per block; x is staged through
// double-buffered LDS with GLOBAL_LOAD_ASYNC_TO_LDS_B128 (no VGPR data path,
// no spills), overlapped with the WMMA stream of the previous chunk.
// ---------------------------------------------------------------------------
__global__ __launch_bounds__(256) void tn_gemm(const float* __restrict__ x,
                                               const float* __restrict__ Bp,
                                               const float* __restrict__ bias,
                                               float* __restrict__ out) {
  __shared__ float lds[2][TM][LDSTRIDE];
  const int tid  = threadIdx.x;
  const int lane = tid & 31;
  const int wave = tid >> 5;
  const size_t sampleBase = (size_t)blockIdx.x * TM;
  const float* xb = x + sampleBase * KDIM;

  v8f acc = {};

#if HAVE_ASYNC_LDS
  // ---- async staging: memory -> LDS without VGPR data movement ----
#define STAGE_CHUNK(CHUNK, BUF)                                                \
  {                                                                            \
    _Pragma("unroll") for (int i = 0; i < 8; ++i) {                            \
      int f = tid + i * 256;                                                   \
      int row = f >> 4, c4 = f & 15;                                           \
      const float* g = xb + (size_t)row * KDIM + (size_t)(CHUNK) * KC + c4 * 4;\
      __builtin_amdgcn_global_load_async_to_lds_b128(                          \
          (gptr_b128)g, (lptr_b128)&lds[BUF][row][c4 * 4], 0, 0);              \
    }                                                                          \
  }

  STAGE_CHUNK(0, 0);
  wait_asynccnt0();
  __syncthreads();

  const int m    = lane & 15;            // A-operand row (sample within tile)
  const int koff = (lane >> 4) << 1;     // lanes 16-31 hold K2/K3

  for (int c = 0; c < NCHUNK; ++c) {
    const int buf = c & 1;
    if (c + 1 < NCHUNK) STAGE_CHUNK(c + 1, buf ^ 1);   // overlap with WMMAs

    const float* arow = &lds[buf][wave * 16 + m][koff];
    const float* bp   = Bp + (size_t)c * WPC * 64 + lane * 2;
#pragma unroll
    for (int i = 0; i < WPC; ++i) {
      v2f av = *(const v2f*)(arow + i * 4);     // ds_load_b64, conflict-free
      v2f bv = *(const v2f*)(bp + i * 64);      // coalesced global b64, L2-hot
      acc = __builtin_amdgcn_wmma_f32_16x16x4_f32(
          false, av, false, bv, (short)0, acc, false, false);
    }
    if (c + 1 < NCHUNK) {
      wait_asynccnt0();      // this wave's asyncs landed in LDS
      __syncthreads();       // => all waves' asyncs landed; old buf free
    }
  }
#undef STAGE_CHUNK

#else
  // ---- fallback: register staging (load -> VGPR -> ds_store) ----
  float4 regs[8];
#pragma unroll
  for (int i = 0; i < 8; ++i) {
    int f = tid + i * 256;
    int row = f >> 4, c4 = f & 15;
    regs[i] = *(const float4*)(xb + (size_t)row * KDIM + c4 * 4);
  }
#pragma unroll
  for (int i = 0; i < 8; ++i) {
    int f = tid + i * 256;
    int row = f >> 4, c4 = f & 15;
    *(float4*)&lds[0][row][c4 * 4] = regs[i];
  }
  __syncthreads();

  const int m    = lane & 15;
  const int koff = (lane >> 4) << 1;

  for (int c = 0; c < NCHUNK; ++c) {
    const int buf = c & 1;
    if (c + 1 < NCHUNK) {
#pragma unroll
      for (int i = 0; i < 8; ++i) {
        int f = tid + i * 256;
        int row = f >> 4, c4 = f & 15;
        regs[i] = *(const float4*)(xb + (size_t)row * KDIM + (c + 1) * KC + c4 * 4);
      }
    }
    const float* arow = &lds[buf][wave * 16 + m][koff];
    const float* bp   = Bp + (size_t)c * WPC * 64 + lane * 2;
#pragma unroll
    for (int i = 0; i < WPC; ++i) {
      v2f av = *(const v2f*)(arow + i * 4);
      v2f bv = *(const v2f*)(bp + i * 64);
      acc = __builtin_amdgcn_wmma_f32_16x16x4_f32(
          false, av, false, bv, (short)0, acc, false, false);
    }
    if (c + 1 < NCHUNK) {
      __syncthreads();
#pragma unroll
      for (int i = 0; i < 8; ++i) {
        int f = tid + i * 256;
        int row = f >> 4, c4 = f & 15;
        *(float4*)&lds[buf ^ 1][row][c4 * 4] = regs[i];
      }
      __syncthreads();
    }
  }
#endif

  // Epilogue: D layout -> VGPR r holds M=r (lanes 0-15) / M=r+8 (lanes 16-31),
  // N = lane%16; only N<8 are real outputs.
  const int n = lane & 15;
  if (n < NOUT) {
    const float b  = bias[n];
    const int   mh = (lane < 16) ? 0 : 8;
#pragma unroll
    for (int r = 0; r < 8; ++r) {
      size_t row = sampleBase + (size_t)wave * 16 + mh + r;
      out[row * NOUT + n] = acc[r] + b;
    }
  }
}

extern "C" void kernel_launch(void* const* d_in, const int* in_sizes, int n_in,
                              void* d_out, int out_size, void* d_ws, size_t ws_size,
                              hipStream_t stream) {
  const float* x    = (const float*)d_in[0];
  const float* n0   = (const float*)d_in[1];
  const float* n1   = (const float*)d_in[2];
  const float* n2   = (const float*)d_in[3];
  const float* bias = (const float*)d_in[4];
  float* out = (float*)d_out;
  float* Bp  = (float*)d_ws;        // 1024 wmma-tiles * 64 floats = 256 KB

  pack_weff<<<16, 256, 0, stream>>>(n0, n1, n2, Bp);
  const int batch = in_sizes[0] / KDIM;          // 65536
  tn_gemm<<<batch / TM, 256, 0, stream>>>(x, Bp, bias, out);
}